// ReallocationMapEncoder_6468220748613
// MI455X (gfx1250) — compile-verified
//
#include <hip/hip_runtime.h>
#include <math.h>

// ---------------------------------------------------------------------------
// ReallocationMapEncoder on MI455X (gfx1250, wave32)
//
// The reference NAC stack is bias-free and activation-free on the data path,
// so the 3->16->16->2 network applied to the affine grid (t/2, a/2048, b/2048)
// collapses to a single 2x3 matrix M = W3*W2*W1 with Wi = tanh(What)*sigm(Mhat):
//     y[t,a,b,k] = gb[k] + (t/2)*M[k,0] + (a/2048)*M[k,1] + (b/2048)*M[k,2]
// Output = 64 MB f32 -> pure store-bandwidth problem (~2.8us @ 23.3 TB/s).
//
// Kernel 1 (1 wave): build W1p/W2/W3p as zero-padded 16x16 f32 tiles in LDS,
//   chain two exact f32 WMMAs (V_WMMA_F32_16X16X4_F32, K=16 via 4 K=4 steps):
//   T = W2@W1p, Mfull = W3p@T. Extract M[2][3], pre-scale, append gb -> d_ws.
// Kernel 2: streaming fill, one (t,a) row (2048*2 f32) per 256-thread block,
//   4 nontemporal b128 stores per thread, 4 v_fma per store.
// ---------------------------------------------------------------------------

#define NSTEPS 2
#define NSYMS  2048

typedef __attribute__((ext_vector_type(2))) float v2f;
typedef __attribute__((ext_vector_type(4))) float f32x4;
typedef __attribute__((ext_vector_type(8))) float v8f;

__device__ __forceinline__ float nac_w(float wh, float mh) {
    // W = tanh(W_hat) * sigmoid(M_hat)
    return tanhf(wh) * (1.0f / (1.0f + expf(-mh)));
}

// ---------------------------------------------------------------------------
// Kernel 1: collapse the NAC stack to 2x3 via exact f32 WMMA. One wave32.
// d_ws layout (8 floats):
//   ws[0..2] = M[0][0]/NSTEPS, M[0][1]/NSYMS, M[0][2]/NSYMS   ws[3] = gb[0]
//   ws[4..6] = M[1][0]/NSTEPS, M[1][1]/NSYMS, M[1][2]/NSYMS   ws[7] = gb[1]
// ---------------------------------------------------------------------------
__global__ __launch_bounds__(32) void rme_collapse_weights(
    const float* __restrict__ gb,
    const float* __restrict__ wh1, const float* __restrict__ mh1,
    const float* __restrict__ wh2, const float* __restrict__ mh2,
    const float* __restrict__ wh3, const float* __restrict__ mh3,
    float* __restrict__ ws8)
{
    __shared__ float W1[16][16];  // 16x3 zero-padded
    __shared__ float W2[16][16];  // 16x16
    __shared__ float W3[16][16];  // 2x16 zero-padded
    __shared__ float T [16][16];  // W2 @ W1

    const int lane = threadIdx.x;          // single wave32

    // Materialize effective NAC weights into LDS (tiny: 3*256 elements).
    for (int idx = lane; idx < 256; idx += 32) {
        const int i = idx >> 4, j = idx & 15;
        W1[i][j] = (j < 3) ? nac_w(wh1[i * 3 + j], mh1[i * 3 + j]) : 0.0f;
        W2[i][j] = nac_w(wh2[idx], mh2[idx]);
        W3[i][j] = (i < 2) ? nac_w(wh3[i * 16 + j], mh3[i * 16 + j]) : 0.0f;
    }
    __syncthreads();

    // f32 WMMA operand layout (ISA 7.12.2, 32-bit A 16x4):
    //   lanes 0-15: VGPR0=K0, VGPR1=K1 ; lanes 16-31: VGPR0=K2, VGPR1=K3
    const int half = lane >> 4;            // which K-pair this lane holds
    const int r    = lane & 15;            // row (A) / column (B)

    // --- T = W2 @ W1 : K=16 as 4 accumulating K=4 WMMA steps ---------------
    v8f acc = {};
#pragma unroll
    for (int k = 0; k < 4; ++k) {
        v2f a, b;
        a.x = W2[r][4 * k + 2 * half];
        a.y = W2[r][4 * k + 2 * half + 1];
        b.x = W1[4 * k + 2 * half][r];
        b.y = W1[4 * k + 2 * half + 1][r];
        acc = __builtin_amdgcn_wmma_f32_16x16x4_f32(
            false, a, false, b, (short)0, acc, false, false);
    }
    // C/D layout: VGPR v -> row v (lanes 0-15) / row v+8 (lanes 16-31), col = lane&15
#pragma unroll
    for (int v = 0; v < 8; ++v)
        T[v + 8 * half][r] = acc[v];
    __syncthreads();

    // --- Mfull = W3 @ T ----------------------------------------------------
    v8f acc2 = {};
#pragma unroll
    for (int k = 0; k < 4; ++k) {
        v2f a, b;
        a.x = W3[r][4 * k + 2 * half];
        a.y = W3[r][4 * k + 2 * half + 1];
        b.x = T[4 * k + 2 * half][r];
        b.y = T[4 * k + 2 * half + 1][r];
        acc2 = __builtin_amdgcn_wmma_f32_16x16x4_f32(
            false, a, false, b, (short)0, acc2, false, false);
    }

    // acc2[0] @ lane n = Mfull[0][n], acc2[1] @ lane n = Mfull[1][n]
    if (lane < 3) {
        const float scale = (lane == 0) ? (1.0f / (float)NSTEPS)
                                        : (1.0f / (float)NSYMS);
        ws8[lane]     = acc2[0] * scale;
        ws8[4 + lane] = acc2[1] * scale;
    } else if (lane == 3) {
        ws8[3] = gb[0];
        ws8[7] = gb[1];
    }
}

// ---------------------------------------------------------------------------
// Kernel 2: bandwidth-bound affine fill of the 2x2048x2048x2 output.
// One block per (t,a) row: 2048 b-values * 2 channels = 4096 f32 = 1024 b128.
// Thread j writes float4 indices {j, j+256, j+512, j+768}: fully coalesced,
// 64 B per thread, nontemporal (nothing re-reads the output).
// ---------------------------------------------------------------------------
__global__ __launch_bounds__(256) void rme_fill(
    const float* __restrict__ ws8, f32x4* __restrict__ out)
{
    const f32x4 c0 = ((const f32x4*)ws8)[0];   // {Mt0, Ma0, Mb0, gb0} (pre-scaled)
    const f32x4 c1 = ((const f32x4*)ws8)[1];   // {Mt1, Ma1, Mb1, gb1}

    const int bid = blockIdx.x;                // 0 .. NSTEPS*NSYMS-1
    const float tf = (float)(bid >> 11);       // t index (scale folded into c.x)
    const float af = (float)(bid & (NSYMS - 1));

    const float base0 = c0.w + tf * c0.x + af * c0.y;   // block-uniform (SALU-able)
    const float base1 = c1.w + tf * c1.x + af * c1.y;
    const float s0 = c0.z, s1 = c1.z;

    f32x4* __restrict__ row = out + (size_t)bid * (NSYMS * 2 / 4);
    const int j = threadIdx.x;
#pragma unroll
    for (int u = 0; u < 4; ++u) {
        const int i  = j + u * 256;            // float4 index within row
        const float b0 = (float)(2 * i);       // covers b = 2i and 2i+1
        const float b1 = (float)(2 * i + 1);
        f32x4 v;
        v.x = base0 + b0 * s0;                 // (b=2i,   k=0)
        v.y = base1 + b0 * s1;                 // (b=2i,   k=1)
        v.z = base0 + b1 * s0;                 // (b=2i+1, k=0)
        v.w = base1 + b1 * s1;                 // (b=2i+1, k=1)
        __builtin_nontemporal_store(v, &row[i]);
    }
}

extern "C" void kernel_launch(void* const* d_in, const int* in_sizes, int n_in,
                              void* d_out, int out_size, void* d_ws, size_t ws_size,
                              hipStream_t stream) {
    // setup_inputs order: 0=market (unused), 1=gb, 2=w_hat1, 3=m_hat1,
    //                     4=w_hat2, 5=m_hat2, 6=w_hat3, 7=m_hat3
    const float* gb  = (const float*)d_in[1];
    const float* wh1 = (const float*)d_in[2];
    const float* mh1 = (const float*)d_in[3];
    const float* wh2 = (const float*)d_in[4];
    const float* mh2 = (const float*)d_in[5];
    const float* wh3 = (const float*)d_in[6];
    const float* mh3 = (const float*)d_in[7];

    float* ws8 = (float*)d_ws;

    rme_collapse_weights<<<1, 32, 0, stream>>>(gb, wh1, mh1, wh2, mh2, wh3, mh3, ws8);

    // 64 MB output: NSTEPS*NSYMS = 4096 row-blocks of 256 threads.
    rme_fill<<<NSTEPS * NSYMS, 256, 0, stream>>>(ws8, (f32x4*)d_out);
}